// SpectralUncertaintyAnalyzer_40132174414096
// MI455X (gfx1250) — compile-verified
//
#include <hip/hip_runtime.h>
#include <math.h>

// ---------------------------------------------------------------------------
// SpectralUncertaintyAnalyzer for gfx1250 (MI455X)
// Truncated 2-D DFT (64x64 corner of rfft2) as two GEMMs per image on
// V_WMMA_F32_16X16X4_F32, with image tiles staged into LDS by the Tensor
// Data Mover (tensor_load_to_lds, TENSORcnt) and double-buffered.
// ---------------------------------------------------------------------------

typedef float v2f __attribute__((ext_vector_type(2)));
typedef float v8f __attribute__((ext_vector_type(8)));
typedef unsigned int u32;
typedef u32 v4u __attribute__((ext_vector_type(4)));
typedef int  v8i __attribute__((ext_vector_type(8)));
typedef int  v4i __attribute__((ext_vector_type(4)));

#define IMG_H 512
#define NMODE 64
#define NIMG 128                            // 32*4 images per tensor
#define MODES_PER_IMG (NMODE * NMODE)       // 4096

// d_out layout (floats, reference tuple order)
#define OFF_E   ((size_t)0)                   // modal_energies        524288
#define OFF_U   ((size_t)524288)              // modal_uncertainties   524288
#define OFF_F   ((size_t)1048576)             // energy_fractions      524288
#define OFF_W   ((size_t)1572864)             // weighted_importance   524288
#define OFF_US  ((size_t)2097152)             // uncertainty_spectrum    4096
#define OFF_ES  ((size_t)2101248)             // energy_spectrum         4096
#define OFF_ERR ((size_t)2105344)             // modal_errors          524288
#define OFF_CAL ((size_t)2629632)             // calibration_scores      4096

// workspace layout (floats)
#define WS_TWT   ((size_t)0)                  // twiddle table 128*512
#define WS_PMRE  ((size_t)65536)              // pred modes re 524288
#define WS_PMIM  ((size_t)(65536 + 524288))
#define WS_GMRE  ((size_t)(65536 + 2*524288))
#define WS_GMIM  ((size_t)(65536 + 3*524288))
#define WS_BLKE  ((size_t)(65536 + 4*524288)) // per-block energy partials 2048
#define WS_TOTE  ((size_t)(65536 + 4*524288 + 2048)) // per-(b,c) totals 128

// TDM pads 2 dwords every 256 dwords -> padded LDS index for linear index i
__device__ __forceinline__ int pad_idx(int i) { return i + 2 * (i >> 8); }
#define IMG_BUF_FLOATS 33024                  // pad_idx(64*512-1)+1 rounded up
#define TS2 68                                // T-tile row stride (bank decorrelation)

__device__ __forceinline__ v8f wmma_f32(v2f a, v2f b, v8f c) {
    // V_WMMA_F32_16X16X4_F32 : D = A(16x4) * B(4x16) + C(16x16), wave32
    return __builtin_amdgcn_wmma_f32_16x16x4_f32(
        false, a, false, b, (short)0, c, false, false);
}

// Issue one TDM transfer: 64x512 f32 tile (row-major, contiguous rows of the
// source image) -> LDS at lds_off, padded +2 dwords per 256 dwords.
__device__ __forceinline__ void tdm_load_chunk(const float* gsrc, u32 lds_off) {
    unsigned long long ga = (unsigned long long)(uintptr_t)gsrc;
    v4u g0;
    g0[0] = 1u;                                        // count=1, user mode
    g0[1] = lds_off;                                   // LDS byte address
    g0[2] = (u32)ga;                                   // global addr lo
    g0[3] = (u32)((ga >> 32) & 0x1FFFFFFu) | (2u << 30); // addr hi | type=2
    v8i g1;
    g1[0] = (int)((2u << 16)      // data_size = 4B
                | (1u << 20)      // pad_enable
                | (7u << 22)      // pad_interval: 256 dwords
                | (1u << 25));    // pad_amount: 2 dwords
    g1[1] = (int)(512u << 16);    // tensor_dim0 = 512 (bits 79:48, low half)
    g1[2] = (int)(512u << 16);    // tensor_dim1 = 512 (bits 111:80, low half)
    g1[3] = (int)(512u << 16);    // tile_dim0 = 512 (bits 127:112)
    g1[4] = 64;                   // tile_dim1 = 64, tile_dim2 = 0
    g1[5] = 512;                  // tensor_dim0_stride = 512
    g1[6] = 0;
    g1[7] = 0;
    v4i gz4 = {0, 0, 0, 0};       // groups 2/3 unused (2-D tile)
    v8i gz8 = {0, 0, 0, 0, 0, 0, 0, 0};
    // clang-23 / therock-10.0 lane: 6-arg form
    __builtin_amdgcn_tensor_load_to_lds(g0, g1, gz4, gz4, gz8, 0);
}

// ---------------------------------------------------------------------------
// Twiddle table, frequency-major: twT[n*512 + y],
//   n in [0,64):   cos(2*pi*n*y/512)
//   n in [64,128): -sin(2*pi*(n-64)*y/512)
// Contiguous in y so every WMMA fragment is a single b64.
// ---------------------------------------------------------------------------
__global__ void twiddle_kernel(float* __restrict__ twT) {
    int i = blockIdx.x * blockDim.x + threadIdx.x;
    if (i >= 128 * 512) return;
    int n = i >> 9;
    int y = i & 511;
    int k = n & 63;
    int r = (y * k) & 511;                   // exact integer phase reduction
    float th = (float)r * (6.2831853071795864769f / 512.0f);
    twT[i] = (n < 64) ? cosf(th) : -sinf(th);
}

// ---------------------------------------------------------------------------
// Per-image truncated DFT. Block = 256 threads (8 waves), one image/block.
//  Stage A (per 64-row x-chunk):  T^T[n,x] = sum_y twT[n,y] * img[x,y]
//     A operand = twiddle rows (global b64), B operand = TDM-staged image
//     chunk in LDS (b64, pad-decorrelated banks). Output -> LDS T tile.
//  Stage B: modes[k1,k2] accumulated over chunks:
//     re += C@T_re + S@T_im,  im += C@T_im - S@T_re   (S stored negated)
// All WMMA-guarding branches are wave-uniform (EXEC all-ones).
// ---------------------------------------------------------------------------
__global__ __launch_bounds__(256) void dft_kernel(
    const float* __restrict__ pred, const float* __restrict__ gt,
    const float* __restrict__ twT,
    float* __restrict__ pmre, float* __restrict__ pmim,
    float* __restrict__ gmre, float* __restrict__ gmim)
{
    __shared__ float imgS[2][IMG_BUF_FLOATS];   // 2 x 129 KB TDM buffers
    __shared__ float T_s[128 * TS2];            // T^T tile: rows n, cols x

    const int tid  = threadIdx.x;
    const int wv   = tid >> 5;        // wave id 0..7 (wave32)
    const int lane = tid & 31;
    const int half = lane >> 4;       // K-pair selector per ISA layout
    const int ln   = lane & 15;

    const int b = blockIdx.x;
    const bool isPred = (b < NIMG);
    const int imgi = isPred ? b : (b - NIMG);
    const float* img = (isPred ? pred : gt) + (size_t)imgi * (IMG_H * IMG_H);
    float* out_re = (isPred ? pmre : gmre) + (size_t)imgi * MODES_PER_IMG;
    float* out_im = (isPred ? pmim : gmim) + (size_t)imgi * MODES_PER_IMG;

    const u32 lds_buf0 = (u32)(uintptr_t)(&imgS[0][0]);
    const u32 lds_buf1 = (u32)(uintptr_t)(&imgS[1][0]);

    // Stage-A role: wave owns freq-row tile mt (16 of 128 rows of T^T)
    const int mt = wv;
    // Stage-B role: waves 0-3 -> modes_re row-tiles, 4-7 -> modes_im
    const int isIm = wv >> 2;
    const int it   = wv & 3;

    // Stage-A A-operand: this wave's twiddle row (contiguous in y)
    const float* arow = twT + (size_t)(mt * 16 + ln) * 512;
    // Stage-B A-operands: cos row k1 and -sin row 64+k1
    const float* crow = twT + (size_t)(it * 16 + ln) * 512;
    const float* nrow = twT + (size_t)(64 + it * 16 + ln) * 512;

    v8f acc2[4];
    #pragma unroll
    for (int j = 0; j < 4; ++j) acc2[j] = (v8f){0,0,0,0,0,0,0,0};

    // Prime the pipeline: DMA chunk 0 into buffer 0 (wave 0 issues; EXEC
    // ignored by tensor ops; branch is wave-uniform).
    if (wv == 0) tdm_load_chunk(img, lds_buf0);

    int buf = 0;
    for (int xc = 0; xc < 8; ++xc) {
        if (wv == 0) {
            if (xc < 7) {
                // prefetch next chunk into the other buffer, then wait for
                // the current chunk (tensor ops complete in order)
                tdm_load_chunk(img + (size_t)(xc + 1) * 64 * IMG_H,
                               (buf == 0) ? lds_buf1 : lds_buf0);
                __builtin_amdgcn_s_wait_tensorcnt(1);
            } else {
                __builtin_amdgcn_s_wait_tensorcnt(0);
            }
        }
        __syncthreads();                       // chunk xc visible to all waves
        const float* ibuf = &imgS[buf][0];

        // ---------------- Stage A: T^T(128x64) = twT(128x512) @ img^T -----
        v8f accA[4];
        #pragma unroll
        for (int nt = 0; nt < 4; ++nt) accA[nt] = (v8f){0,0,0,0,0,0,0,0};

        for (int ks = 0; ks < IMG_H; ks += 4) {
            const int ka = ks + 2 * half;
            const v2f A = *(const v2f*)(arow + ka);          // global b64
            #pragma unroll
            for (int nt = 0; nt < 4; ++nt) {
                const int x = nt * 16 + ln;                  // spatial row
                const v2f Bf = *(const v2f*)(ibuf + pad_idx(x * IMG_H + ka));
                accA[nt] = wmma_f32(A, Bf, accA[nt]);
            }
        }
        #pragma unroll
        for (int nt = 0; nt < 4; ++nt) {
            #pragma unroll
            for (int r = 0; r < 8; ++r) {
                const int row = mt * 16 + (half ? (8 + r) : r);
                T_s[row * TS2 + nt * 16 + ln] = accA[nt][r];
            }
        }
        __syncthreads();

        // ---------------- Stage B: accumulate 64x64 complex modes ---------
        const int row0 = xc * 64;
        for (int ks = 0; ks < 64; ks += 4) {
            const int xg = row0 + ks + 2 * half;
            const v2f Ac = *(const v2f*)(crow + xg);
            const v2f An = *(const v2f*)(nrow + xg);
            const int xr = ks + 2 * half;
            #pragma unroll
            for (int j = 0; j < 4; ++j) {
                const int k2 = j * 16 + ln;
                const v2f Bre = *(const v2f*)(&T_s[k2 * TS2 + xr]);
                const v2f Bim = *(const v2f*)(&T_s[(64 + k2) * TS2 + xr]);
                if (!isIm) {
                    // re += C*T_re + S*T_im = C*T_re + (-S)*(-T_im)
                    acc2[j] = wmma_f32(Ac, Bre, acc2[j]);
                    v2f Bn; Bn[0] = -Bim[0]; Bn[1] = -Bim[1];
                    acc2[j] = wmma_f32(An, Bn, acc2[j]);
                } else {
                    // im += C*T_im - S*T_re = C*T_im + (-S)*T_re
                    acc2[j] = wmma_f32(Ac, Bim, acc2[j]);
                    acc2[j] = wmma_f32(An, Bre, acc2[j]);
                }
            }
        }
        __syncthreads();                       // T_s / imgS[buf] reusable
        buf ^= 1;
    }

    float* outp = isIm ? out_im : out_re;
    #pragma unroll
    for (int j = 0; j < 4; ++j) {
        #pragma unroll
        for (int r = 0; r < 8; ++r) {
            const int row = it * 16 + (half ? (8 + r) : r);
            outp[row * 64 + j * 16 + ln] = acc2[j][r];
        }
    }
}

// ---------------------------------------------------------------------------
// Per-mode MLP (2->64 relu ->32 relu ->1 softplus) fused with energies/errors.
// ---------------------------------------------------------------------------
__global__ __launch_bounds__(256) void mlp_kernel(
    const float* __restrict__ pmre, const float* __restrict__ pmim,
    const float* __restrict__ gmre, const float* __restrict__ gmim,
    const float* __restrict__ W1, const float* __restrict__ b1,
    const float* __restrict__ W2, const float* __restrict__ b2,
    const float* __restrict__ W3, const float* __restrict__ b3,
    float* __restrict__ out, float* __restrict__ blkE)
{
    __shared__ float sW1[128], sb1[64], sW2[2048], sb2[32], sW3[32], sb3;
    __shared__ float red[256];
    const int tid = threadIdx.x;

    for (int t = tid; t < 128;  t += 256) sW1[t] = W1[t];
    for (int t = tid; t < 64;   t += 256) sb1[t] = b1[t];
    for (int t = tid; t < 2048; t += 256) sW2[t] = W2[t];
    if (tid < 32) { sb2[tid] = b2[tid]; sW3[tid] = W3[tid]; }
    if (tid == 0) sb3 = b3[0];
    __syncthreads();

    const size_t idx = (size_t)blockIdx.x * 256 + tid;   // < 524288
    const float mre = pmre[idx], mim = pmim[idx];
    const float gre = gmre[idx], gim = gmim[idx];
    const float energy = mre * mre + mim * mim;
    const float dre = mre - gre, dim2 = mim - gim;
    const float err = dre * dre + dim2 * dim2;

    float acc[32];
    #pragma unroll
    for (int k = 0; k < 32; ++k) acc[k] = sb2[k];
    for (int j = 0; j < 64; ++j) {
        float h = fmaf(mre, sW1[j], fmaf(mim, sW1[64 + j], sb1[j]));
        h = fmaxf(h, 0.0f);
        #pragma unroll
        for (int k = 0; k < 32; ++k) acc[k] = fmaf(h, sW2[j * 32 + k], acc[k]);
    }
    float o = sb3;
    #pragma unroll
    for (int k = 0; k < 32; ++k) o = fmaf(fmaxf(acc[k], 0.0f), sW3[k], o);
    const float u = (o > 0.0f) ? (o + log1pf(expf(-o))) : log1pf(expf(o));

    out[OFF_E + idx]   = energy;
    out[OFF_U + idx]   = u;
    out[OFF_ERR + idx] = err;

    red[tid] = energy;
    __syncthreads();
    #pragma unroll
    for (int s = 128; s > 0; s >>= 1) {
        if (tid < s) red[tid] += red[tid + s];
        __syncthreads();
    }
    if (tid == 0) blkE[blockIdx.x] = red[0];   // 16 blocks per (b,c)
}

// Fixed-order reduction of the 16 per-block partials per (b,c) -> totE[128]
__global__ void tot_kernel(const float* __restrict__ blkE, float* __restrict__ totE) {
    const int bc = threadIdx.x;
    if (bc < NIMG) {
        float s = 0.0f;
        #pragma unroll
        for (int t = 0; t < 16; ++t) s += blkE[bc * 16 + t];
        totE[bc] = s;
    }
}

__global__ __launch_bounds__(256) void frac_kernel(
    float* __restrict__ out, const float* __restrict__ totE)
{
    const size_t idx = (size_t)blockIdx.x * 256 + threadIdx.x;
    const int bc = (int)(idx >> 12);
    const float e = out[OFF_E + idx];
    const float f = e / (totE[bc] + 1e-8f);
    out[OFF_F + idx] = f;
    out[OFF_W + idx] = f * out[OFF_U + idx];
}

// Spectra + per-mode Pearson correlation over the 128 (b,c) samples.
__global__ __launch_bounds__(256) void calib_kernel(float* __restrict__ out)
{
    const int m = blockIdx.x * 256 + threadIdx.x;   // 0..4095
    float Su = 0.f, Se = 0.f, Sen = 0.f, Suu = 0.f, See = 0.f, Sue = 0.f;
    for (int bc = 0; bc < NIMG; ++bc) {
        const size_t o = (size_t)bc * MODES_PER_IMG + m;
        const float u  = out[OFF_U + o];
        const float e  = out[OFF_ERR + o];
        const float en = out[OFF_E + o];
        Su += u; Se += e; Sen += en;
        Suu = fmaf(u, u, Suu);
        See = fmaf(e, e, See);
        Sue = fmaf(u, e, Sue);
    }
    const float invn = 1.0f / 128.0f;
    out[OFF_US + m] = Su * invn;
    out[OFF_ES + m] = Sen * invn;
    const float num = Sue - Su * Se * invn;
    const float va  = fmaxf(Suu - Su * Su * invn, 0.0f);
    const float vb  = fmaxf(See - Se * Se * invn, 0.0f);
    out[OFF_CAL + m] = num / (sqrtf(va * vb) + 1e-8f);
}

// ---------------------------------------------------------------------------
extern "C" void kernel_launch(void* const* d_in, const int* in_sizes, int n_in,
                              void* d_out, int out_size, void* d_ws, size_t ws_size,
                              hipStream_t stream) {
    const float* pred = (const float*)d_in[0];
    // d_in[1] = uncertainty (unused by the reference computation)
    const float* gt  = (const float*)d_in[2];
    const float* W1  = (const float*)d_in[3];
    const float* b1  = (const float*)d_in[4];
    const float* W2  = (const float*)d_in[5];
    const float* b2  = (const float*)d_in[6];
    const float* W3  = (const float*)d_in[7];
    const float* b3  = (const float*)d_in[8];
    float* out = (float*)d_out;
    float* ws  = (float*)d_ws;

    float* twT  = ws + WS_TWT;
    float* pmre = ws + WS_PMRE;
    float* pmim = ws + WS_PMIM;
    float* gmre = ws + WS_GMRE;
    float* gmim = ws + WS_GMIM;
    float* blkE = ws + WS_BLKE;
    float* totE = ws + WS_TOTE;

    twiddle_kernel<<<256, 256, 0, stream>>>(twT);
    dft_kernel<<<256, 256, 0, stream>>>(pred, gt, twT, pmre, pmim, gmre, gmim);
    mlp_kernel<<<2048, 256, 0, stream>>>(pmre, pmim, gmre, gmim,
                                         W1, b1, W2, b2, W3, b3, out, blkE);
    tot_kernel<<<1, 128, 0, stream>>>(blkE, totE);
    frac_kernel<<<2048, 256, 0, stream>>>(out, totE);
    calib_kernel<<<16, 256, 0, stream>>>(out);
}